// MultiResolutionHashEncoding_644245095035
// MI455X (gfx1250) — compile-verified
//
#include <hip/hip_runtime.h>
#include <math.h>
#include <stdint.h>

#define TABLE_SIZE 524288u
#define NUM_LEVELS 16
#define FEAT_STRIDE (TABLE_SIZE * NUM_LEVELS) /* 8388608 floats between feature planes */
#define PTS_PER_BLOCK 16
#define BLOCK 256

typedef float f32x2 __attribute__((ext_vector_type(2)));

struct Scalings { float s[NUM_LEVELS]; };

__device__ __forceinline__ float sel3(float c0, float c1, float c2, int m) {
    // dynamic 3-way select without scratch-backed arrays (v_cndmask chain)
    return (m == 0) ? c0 : ((m == 1) ? c1 : c2);
}

__global__ __launch_bounds__(BLOCK) void hashenc_kernel(
    const float* __restrict__ x,
    const float* __restrict__ table,
    float* __restrict__ out,
    const Scalings S, const int n_points)
{
    __shared__ float shx[PTS_PER_BLOCK * 3]; // 48 coords for this block's points
    __shared__ float shS[NUM_LEVELS];        // per-level scales (dynamic-index friendly)

    const int t = threadIdx.x;
    const int pBase = blockIdx.x * PTS_PER_BLOCK;

    if (t < NUM_LEVELS) shS[t] = S.s[t];

    // CDNA5 async global->LDS staging of the block's 48 coordinate floats
    // (ASYNCcnt-tracked; avoids 16x redundant per-lane coordinate loads).
    if (t < PTS_PER_BLOCK * 3) {
        uint32_t lds_addr = (uint32_t)(uintptr_t)&shx[t];       // low 32b = LDS offset
        uint32_t voff     = ((uint32_t)(pBase * 3 + t)) * 4u;   // byte offset from saddr
        asm volatile("global_load_async_to_lds_b32 %0, %1, %2"
                     :: "v"(lds_addr), "v"(voff), "s"(x)
                     : "memory");
    }
    asm volatile("s_wait_asynccnt 0" ::: "memory");
    __syncthreads();

    const int j  = t & 15;   // level (lanes 0..15 of a wave = one point's 16 levels)
    const int pl = t >> 4;   // point within block
    const int p  = pBase + pl;
    if (p >= n_points) return;

    const float c0 = shx[pl * 3 + 0];
    const float c1 = shx[pl * 3 + 1];
    const float c2 = shx[pl * 3 + 2];

    // ---- own-level lattice coords -> 8 corner hashes (exact int32-wrap semantics)
    const float s  = shS[j];
    const float sx = c0 * s, sy = c1 * s, sz = c2 * s;
    const uint32_t P1 = 2654435761u, P2 = 805459861u;
    const uint32_t hxf = (uint32_t)(int)floorf(sx);
    const uint32_t hxc = (uint32_t)(int)ceilf(sx);
    const uint32_t hyf = (uint32_t)(int)floorf(sy) * P1;
    const uint32_t hyc = (uint32_t)(int)ceilf(sy)  * P1;
    const uint32_t hzf = (uint32_t)(int)floorf(sz) * P2;
    const uint32_t hzc = (uint32_t)(int)ceilf(sz)  * P2;

    const uint32_t lvl = (uint32_t)j << 19;      // + level*TABLE_SIZE
    const uint32_t M   = TABLE_SIZE - 1u;        // py-mod of pow2 == bitwise AND
    const uint32_t h0 = (((hxc ^ hyc) ^ hzc) & M) + lvl; // (c,c,c)
    const uint32_t h1 = (((hxc ^ hyc) ^ hzf) & M) + lvl; // (c,c,f)
    const uint32_t h2 = (((hxc ^ hyf) ^ hzc) & M) + lvl; // (c,f,c)
    const uint32_t h3 = (((hxf ^ hyc) ^ hzc) & M) + lvl; // (f,c,c)
    const uint32_t h4 = (((hxc ^ hyf) ^ hzf) & M) + lvl; // (c,f,f)
    const uint32_t h5 = (((hxf ^ hyc) ^ hzf) & M) + lvl; // (f,c,f)
    const uint32_t h6 = (((hxf ^ hyf) ^ hzc) & M) + lvl; // (f,f,c)
    const uint32_t h7 = (((hxf ^ hyf) ^ hzf) & M) + lvl; // (f,f,f)

    // ---- 16 independent gathers, batched (table is L2-resident: 64MB < 192MB L2)
    const float* __restrict__ t0 = table;
    const float* __restrict__ t1 = table + FEAT_STRIDE;
    const float a0 = t0[h0], a1 = t0[h1], a2 = t0[h2], a3 = t0[h3];
    const float a4 = t0[h4], a5 = t0[h5], a6 = t0[h6], a7 = t0[h7];
    const float b0 = t1[h0], b1 = t1[h1], b2 = t1[h2], b3 = t1[h3];
    const float b4 = t1[h4], b5 = t1[h5], b6 = t1[h6], b7 = t1[h7];

    // ---- lerp weights: reference reshape(N,16,3)->(N,3,16) scramble:
    // weight k of level j = frac at flat index (16k + j) of level-major array
    const int ka = j, kb = j + 16, kc = j + 32;
    const float va = sel3(c0, c1, c2, ka % 3) * shS[ka / 3];
    const float vb = sel3(c0, c1, c2, kb % 3) * shS[kb / 3];
    const float vc = sel3(c0, c1, c2, kc % 3) * shS[kc / 3];
    const float px = va - floorf(va);
    const float py = vb - floorf(vb);
    const float pz = vc - floorf(vc);
    const float qx = 1.0f - px, qy = 1.0f - py, qz = 1.0f - pz;

    const float A03 = a0 * px + a3 * qx;
    const float A12 = a1 * px + a2 * qx;
    const float A56 = a5 * px + a6 * qx;
    const float A47 = a4 * px + a7 * qx;
    const float AA  = A03 * py + A12 * qy;
    const float AB  = A47 * py + A56 * qy;
    const float e0  = AA * pz + AB * qz;

    const float B03 = b0 * px + b3 * qx;
    const float B12 = b1 * px + b2 * qx;
    const float B56 = b5 * px + b6 * qx;
    const float B47 = b4 * px + b7 * qx;
    const float BA  = B03 * py + B12 * qy;
    const float BB  = B47 * py + B56 * qy;
    const float e1  = BA * pz + BB * qz;

    // 8B per lane, 256B contiguous per wave; NT so the 128MB stream
    // does not evict the L2-resident hash table.
    f32x2 r; r.x = e0; r.y = e1;
    __builtin_nontemporal_store(r, (f32x2*)(out + ((size_t)p * 32 + (size_t)(j * 2))));
}

extern "C" void kernel_launch(void* const* d_in, const int* in_sizes, int n_in,
                              void* d_out, int out_size, void* d_ws, size_t ws_size,
                              hipStream_t stream) {
    const float* x     = (const float*)d_in[0];
    const float* table = (const float*)d_in[1];
    float* out         = (float*)d_out;
    const int n_points = in_sizes[0] / 3;

    // SCALINGS exactly as numpy (double precision): floor(16 * growth^l)
    Scalings S;
    const double growth = exp((log(2048.0) - log(16.0)) / 15.0);
    for (int l = 0; l < NUM_LEVELS; ++l)
        S.s[l] = (float)floor(16.0 * pow(growth, (double)l));

    const int blocks = (n_points + PTS_PER_BLOCK - 1) / PTS_PER_BLOCK;
    hipLaunchKernelGGL(hashenc_kernel, dim3(blocks), dim3(BLOCK), 0, stream,
                       x, table, out, S, n_points);
}